// Decoder_61813169324478
// MI455X (gfx1250) — compile-verified
//
#include <hip/hip_runtime.h>

// Decoder step: attention + LSTM + vocab projection, f32 end-to-end.
// GEMMs: V_WMMA_F32_16X16X4_F32 (wave32), A-tiles staged into LDS with
// double-buffered GLOBAL_LOAD_ASYNC_TO_LDS_B128 (ASYNCcnt-tracked).

typedef float v2f __attribute__((ext_vector_type(2)));
typedef float v4f __attribute__((ext_vector_type(4)));
typedef float v8f __attribute__((ext_vector_type(8)));
typedef int   v4i __attribute__((vector_size(16)));   // matches builtin's V4i

#define NSEQ 256
#define SLEN 128
#define H2   1024   // 2*H
#define HDIM 512
#define EDIM 256
#define KEXT 1792   // 2H + E + H
#define G4H  2048   // 4*H
#define OVOC 32000

#define LDSP 36     // LDS row pitch (floats): 144B = 16B-aligned rows, no bank conflicts

#define AS1 __attribute__((address_space(1)))
#define AS3 __attribute__((address_space(3)))

static __device__ __forceinline__ v8f wmma4(v2f a, v2f b, v8f c) {
    // D = A(16x4) * B(4x16) + C, f32 accumulate
    return __builtin_amdgcn_wmma_f32_16x16x4_f32(
        /*neg_a=*/false, a, /*neg_b=*/false, b,
        /*c_mod=*/(short)0, c, /*reuse_a=*/false, /*reuse_b=*/false);
}

// 16B global -> LDS copy, async if the CDNA5 builtin is available.
static __device__ __forceinline__ void async_copy16(const float* g, float* l) {
#if __has_builtin(__builtin_amdgcn_global_load_async_to_lds_b128)
    __builtin_amdgcn_global_load_async_to_lds_b128(
        (AS1 v4i*)(AS1 void*)(void*)g,
        (AS3 v4i*)(AS3 void*)(void*)l, 0, 0);
#else
    *(v4f*)l = *(const v4f*)g;   // sync fallback (same barrier structure is correct)
#endif
}

static __device__ __forceinline__ void wait_async0() {
#if __has_builtin(__builtin_amdgcn_s_wait_asynccnt)
    __builtin_amdgcn_s_wait_asynccnt(0);
#else
    asm volatile("s_wait_asynccnt 0" ::: "memory");
#endif
}

// Stage a 256x32 f32 chunk of A (row-major, leading dim ldA) into an LDS
// buffer with LDSP pitch. 8 x b128 per thread, 256 threads.
static __device__ __forceinline__ void stage_chunk(
    const float* __restrict__ A, int ldA, int kc, float* __restrict__ dst, int tid)
{
    #pragma unroll
    for (int it = 0; it < 8; ++it) {
        const int idx = it * 256 + tid;   // 0..2047 float4 slots
        const int r   = idx >> 3;         // row 0..255
        const int c4  = idx & 7;          // float4 within 32-float chunk
        async_copy16(A + (size_t)r * ldA + kc + c4 * 4, dst + r * LDSP + c4 * 4);
    }
}

// ---------------------------------------------------------------------------
// Kernel 1: attention energies -> softmax -> context; build lstm_ext row
//   lstm_ext[n] = [ context(1024) | emb[x[n]](256) | h[n](512) ]
// One block (8 waves) per n. enc slice (512KB) is read twice; second pass
// hits the 192MB L2.
// ---------------------------------------------------------------------------
__global__ __launch_bounds__(256) void k_attn(
    const int* __restrict__ x, const float* __restrict__ enc,
    const float* __restrict__ hid, const float* __restrict__ emb,
    const float* __restrict__ We, const float* __restrict__ be,
    float* __restrict__ lstm_ext)
{
    const int n    = blockIdx.x;
    const int tid  = threadIdx.x;
    const int lane = tid & 31;
    const int wave = tid >> 5;

    __shared__ __align__(16) float we_e[H2];   // W_energy[512:1536]
    __shared__ __align__(16) float h_s[HDIM];
    __shared__ float energy[SLEN];
    __shared__ float red[8];
    __shared__ float hdot_s;

    const float* h = hid + (size_t)n * HDIM;

    for (int i = tid; i < H2; i += 256) we_e[i] = We[HDIM + i];

    // hdot = dot(h, W_energy[0:512])  (constant over s)
    float p = 0.f;
    for (int i = tid; i < HDIM; i += 256) {
        float hv = h[i];
        h_s[i] = hv;
        p += hv * We[i];
    }
    #pragma unroll
    for (int off = 16; off; off >>= 1) p += __shfl_xor(p, off, 32);
    if (lane == 0) red[wave] = p;
    __syncthreads();
    if (tid == 0) {
        float s = 0.f;
        #pragma unroll
        for (int i = 0; i < 8; ++i) s += red[i];
        hdot_s = s + be[0];
    }
    __syncthreads();
    const float hdot = hdot_s;

    // energies: wave w handles rows s = w*16 .. w*16+15
    const float* encn = enc + (size_t)n * SLEN * H2;
    for (int t = 0; t < 16; ++t) {
        const int s = wave * 16 + t;
        const float* row = encn + (size_t)s * H2;
        float sum = 0.f;
        #pragma unroll
        for (int j = 0; j < 8; ++j) {
            const int o = (j * 32 + lane) * 4;
            v4f ev = *(const v4f*)(row + o);
            v4f wv = *(const v4f*)(we_e + o);
            sum += ev.x * wv.x + ev.y * wv.y + ev.z * wv.z + ev.w * wv.w;
        }
        #pragma unroll
        for (int off = 16; off; off >>= 1) sum += __shfl_xor(sum, off, 32);
        if (lane == 0) {
            float e = sum + hdot;
            energy[s] = e > 0.f ? e : 0.f;   // relu
        }
    }
    __syncthreads();

    // softmax over 128 (wave 0)
    if (wave == 0) {
        float e0 = energy[lane], e1 = energy[lane + 32];
        float e2 = energy[lane + 64], e3 = energy[lane + 96];
        float m = fmaxf(fmaxf(e0, e1), fmaxf(e2, e3));
        #pragma unroll
        for (int off = 16; off; off >>= 1) m = fmaxf(m, __shfl_xor(m, off, 32));
        float x0 = __expf(e0 - m), x1 = __expf(e1 - m);
        float x2 = __expf(e2 - m), x3 = __expf(e3 - m);
        float sum = x0 + x1 + x2 + x3;
        #pragma unroll
        for (int off = 16; off; off >>= 1) sum += __shfl_xor(sum, off, 32);
        const float inv = 1.f / sum;
        energy[lane]      = x0 * inv;
        energy[lane + 32] = x1 * inv;
        energy[lane + 64] = x2 * inv;
        energy[lane + 96] = x3 * inv;
    }
    __syncthreads();

    // context: thread owns 4 columns (second pass over enc slice -> L2 hits)
    const int c0 = tid * 4;
    v4f acc; acc.x = 0.f; acc.y = 0.f; acc.z = 0.f; acc.w = 0.f;
    for (int s = 0; s < SLEN; ++s) {
        const float a = energy[s];
        v4f ev = *(const v4f*)(encn + (size_t)s * H2 + c0);
        acc.x += a * ev.x; acc.y += a * ev.y;
        acc.z += a * ev.z; acc.w += a * ev.w;
    }
    float* dst = lstm_ext + (size_t)n * KEXT;
    *(v4f*)(dst + c0) = acc;

    // embedding gather
    const int tok = x[n];
    dst[H2 + tid] = emb[(size_t)tok * EDIM + tid];

    // h copy
    dst[H2 + EDIM + 2 * tid]     = h_s[2 * tid];
    dst[H2 + EDIM + 2 * tid + 1] = h_s[2 * tid + 1];
}

// ---------------------------------------------------------------------------
// Kernel 2: gates = lstm_ext(256x1792) @ [W_ih|W_hh]^T (2048x1792) + biases
// grid 16 blocks x 8 waves; wave = one 16-col tile, all 16 row tiles (acc[16]).
// A chunks (256x32) double-buffered in LDS via async-to-LDS copies.
// ---------------------------------------------------------------------------
__global__ __launch_bounds__(256) void k_gates(
    const float* __restrict__ A, const float* __restrict__ Wih,
    const float* __restrict__ Whh, const float* __restrict__ bih,
    const float* __restrict__ bhh, float* __restrict__ gates)
{
    const int tid = threadIdx.x;
    const int lane = tid & 31;
    const int lane15 = lane & 15;
    const int laneHi = lane >> 4;
    const int wave = tid >> 5;
    const int n0 = blockIdx.x * 128 + wave * 16;

    __shared__ __align__(16) float As[2][256 * LDSP];

    v8f acc[16];
    #pragma unroll
    for (int mt = 0; mt < 16; ++mt)
        #pragma unroll
        for (int r = 0; r < 8; ++r) acc[mt][r] = 0.f;

    const int nchunks = KEXT / 32;          // 56
    stage_chunk(A, KEXT, 0, As[0], tid);    // prologue

    for (int c = 0; c < nchunks; ++c) {
        const int kc = c * 32;
        const float* Bsrc; int ldb, kk;
        if (kc < H2 + EDIM) { Bsrc = Wih; ldb = H2 + EDIM; kk = kc; }
        else                { Bsrc = Whh; ldb = HDIM;      kk = kc - (H2 + EDIM); }

        wait_async0();       // my chunk-c copies landed in LDS
        __syncthreads();     // everyone's landed; everyone done reading other buf
        if (c + 1 < nchunks)
            stage_chunk(A, KEXT, kc + 32, As[(c + 1) & 1], tid);

        const float* Sa = As[c & 1];
        const float* Brow = Bsrc + (size_t)(n0 + lane15) * ldb + kk + 2 * laneHi;
        #pragma unroll
        for (int k = 0; k < 32; k += 4) {
            v2f bv = *(const v2f*)(Brow + k);
            #pragma unroll
            for (int mt = 0; mt < 16; ++mt) {
                v2f av = *(const v2f*)(Sa + (mt * 16 + lane15) * LDSP + k + 2 * laneHi);
                acc[mt] = wmma4(av, bv, acc[mt]);
            }
        }
    }

    const int col = n0 + lane15;
    const float bias = bih[col] + bhh[col];
    #pragma unroll
    for (int mt = 0; mt < 16; ++mt)
        #pragma unroll
        for (int r = 0; r < 8; ++r) {
            const int row = mt * 16 + r + laneHi * 8;
            gates[(size_t)row * G4H + col] = acc[mt][r] + bias;
        }
}

// ---------------------------------------------------------------------------
// Kernel 3: LSTM elementwise -> h_new, c_new (into d_out tail)
// ---------------------------------------------------------------------------
__global__ __launch_bounds__(256) void k_lstm(
    const float* __restrict__ gates, const float* __restrict__ cell,
    float* __restrict__ hnew, float* __restrict__ cnew)
{
    const int idx = blockIdx.x * 256 + threadIdx.x;   // 0..131071
    const int nrow = idx >> 9;
    const int hh   = idx & 511;
    const float* g = gates + (size_t)nrow * G4H;
    const float gi = g[hh], gf = g[hh + 512], gg = g[hh + 1024], go = g[hh + 1536];
    const float i = 1.f / (1.f + __expf(-gi));
    const float f = 1.f / (1.f + __expf(-gf));
    const float o = 1.f / (1.f + __expf(-go));
    const float gt = tanhf(gg);
    const float c = f * cell[idx] + i * gt;
    const float h = o * tanhf(c);
    hnew[idx] = h;
    cnew[idx] = c;
}

// ---------------------------------------------------------------------------
// Kernel 4: predictions = h_new(256x512) @ W_out^T (32000x512) + b_out
// grid 250 blocks x 8 waves; wave = one 16-col tile, all 16 row tiles.
// W_out streamed exactly once (16x B-frag reuse); A double-buffered async.
// ---------------------------------------------------------------------------
__global__ __launch_bounds__(256) void k_out(
    const float* __restrict__ A, const float* __restrict__ W,
    const float* __restrict__ bout, float* __restrict__ out)
{
    const int tid = threadIdx.x;
    const int lane = tid & 31;
    const int lane15 = lane & 15;
    const int laneHi = lane >> 4;
    const int wave = tid >> 5;
    const int n0 = blockIdx.x * 128 + wave * 16;

    __shared__ __align__(16) float As[2][256 * LDSP];

    v8f acc[16];
    #pragma unroll
    for (int mt = 0; mt < 16; ++mt)
        #pragma unroll
        for (int r = 0; r < 8; ++r) acc[mt][r] = 0.f;

    const float* Wrow = W + (size_t)(n0 + lane15) * HDIM + 2 * laneHi;

    const int nchunks = HDIM / 32;          // 16
    stage_chunk(A, HDIM, 0, As[0], tid);    // prologue

    for (int c = 0; c < nchunks; ++c) {
        const int kc = c * 32;

        wait_async0();
        __syncthreads();
        if (c + 1 < nchunks)
            stage_chunk(A, HDIM, kc + 32, As[(c + 1) & 1], tid);

        // prefetch next W_out chunk (global_prefetch_b8)
        if (kc + 32 < HDIM) __builtin_prefetch(Wrow + kc + 32, 0, 1);

        const float* Sa = As[c & 1];
        #pragma unroll
        for (int k = 0; k < 32; k += 4) {
            v2f bv = *(const v2f*)(Wrow + kc + k);
            #pragma unroll
            for (int mt = 0; mt < 16; ++mt) {
                v2f av = *(const v2f*)(Sa + (mt * 16 + lane15) * LDSP + k + 2 * laneHi);
                acc[mt] = wmma4(av, bv, acc[mt]);
            }
        }
    }

    const int col = n0 + lane15;
    const float bias = bout[col];
    #pragma unroll
    for (int mt = 0; mt < 16; ++mt)
        #pragma unroll
        for (int r = 0; r < 8; ++r) {
            const int row = mt * 16 + r + laneHi * 8;
            out[(size_t)row * OVOC + col] = acc[mt][r] + bias;
        }
}

extern "C" void kernel_launch(void* const* d_in, const int* in_sizes, int n_in,
                              void* d_out, int out_size, void* d_ws, size_t ws_size,
                              hipStream_t stream) {
    const int*   x    = (const int*)  d_in[0];
    const float* enc  = (const float*)d_in[1];
    const float* hid  = (const float*)d_in[2];
    const float* cell = (const float*)d_in[3];
    const float* emb  = (const float*)d_in[4];
    const float* We   = (const float*)d_in[5];
    const float* be   = (const float*)d_in[6];
    const float* Wih  = (const float*)d_in[7];
    const float* Whh  = (const float*)d_in[8];
    const float* bih  = (const float*)d_in[9];
    const float* bhh  = (const float*)d_in[10];
    const float* Wout = (const float*)d_in[11];
    const float* bout = (const float*)d_in[12];

    float* out  = (float*)d_out;
    float* pred = out;                                   // 256*32000
    float* hnew = out + (size_t)NSEQ * OVOC;             // 256*512
    float* cnew = hnew + (size_t)NSEQ * HDIM;            // 256*512

    float* ws       = (float*)d_ws;
    float* lstm_ext = ws;                                // 256*1792
    float* gates    = ws + (size_t)NSEQ * KEXT;          // 256*2048

    k_attn <<<NSEQ, 256, 0, stream>>>(x, enc, hid, emb, We, be, lstm_ext);
    k_gates<<<G4H / 128, 256, 0, stream>>>(lstm_ext, Wih, Whh, bih, bhh, gates);
    k_lstm <<<(NSEQ * HDIM) / 256, 256, 0, stream>>>(gates, cell, hnew, cnew);
    k_out  <<<OVOC / 128, 256, 0, stream>>>(hnew, Wout, bout, pred);
}